// BidirectionalGNN_24498493456395
// MI455X (gfx1250) — compile-verified
//
#include <hip/hip_runtime.h>

typedef __attribute__((ext_vector_type(2))) float v2f;
typedef __attribute__((ext_vector_type(8))) float v8f;

#define DIN   256
#define HID   256
#define HC    128   // HEADS * C
#define HEADS 4
#define CH    32
#define NEG_SLOPE 0.2f
#define LDSP  260   // LDS row pitch in floats (260 % 64 == 4 -> conflict-free, 1040B = 16B-aligned)

// ---- ordered-int encoding for float atomic max (monotonic map) ----
__device__ __forceinline__ int enc_f(float v) {
    int i = __float_as_int(v);
    return i >= 0 ? i : (i ^ 0x7FFFFFFF);
}
__device__ __forceinline__ float dec_f(int j) {
    return __int_as_float(j >= 0 ? j : (j ^ 0x7FFFFFFF));
}

// ---- CDNA5 async copy: global -> LDS (ASYNCcnt-tracked) ----
__device__ __forceinline__ void async_g2l_b128(const float* gaddr, const float* lds_generic) {
    // low 32 bits of a generic pointer to __shared__ is the LDS byte offset
    unsigned lds = (unsigned)(unsigned long long)lds_generic;
    asm volatile("global_load_async_to_lds_b128 %0, %1, off"
                 :: "v"(lds), "v"(gaddr)
                 : "memory");
}
__device__ __forceinline__ void async_wait0() {
    asm volatile("s_wait_asynccnt 0x0" ::: "memory");
}

// ---------------- init: agg=0, den=0, m=INT_MIN ----------------
__global__ void k_init(float* __restrict__ agg, float* __restrict__ den,
                       int* __restrict__ m, long nAgg, long nDM) {
    long i = (long)blockIdx.x * blockDim.x + threadIdx.x;
    long stride = (long)gridDim.x * blockDim.x;
    for (long t = i; t < nAgg; t += stride) agg[t] = 0.0f;
    for (long t = i; t < nDM; t += stride) { den[t] = 0.0f; m[t] = (int)0x80000000; }
}

// -------- dual projection GEMM: xlF = x@W_f, xlB = x@W_b (fp32 WMMA) --------
// One block = one 16-row M tile. A tile async-staged to LDS once; each of the
// 8 waves computes one 16-col tile of BOTH directions off shared A fragments.
__global__ void __launch_bounds__(256)
k_proj_wmma(const float* __restrict__ x,
            const float* __restrict__ Wf,
            const float* __restrict__ Wb,
            float* __restrict__ xlF, float* __restrict__ xlB) {
    __shared__ float xs[16 * LDSP];
    const int tid  = threadIdx.x;
    const int lane = tid & 31;
    const int wave = tid >> 5;
    const int mtile = blockIdx.x;

    // stage 16 x 256 floats (16KB) : 1024 16B chunks, 4 per thread
#pragma unroll
    for (int i = 0; i < 4; ++i) {
        int c    = tid + 256 * i;
        int row  = c >> 6;            // 64 chunks per 256-float row
        int kpos = (c & 63) << 2;     // float offset within row
        const float* g = x + (long)(mtile * 16 + row) * DIN + kpos;
        async_g2l_b128(g, &xs[row * LDSP + kpos]);
    }
    async_wait0();
    __syncthreads();

    const int l15   = lane & 15;
    const int khalf = (lane >> 4) << 1;          // 0 or 2
    const int ncol  = wave * 16 + l15;           // 8 waves cover 128 cols
    const float* bfp  = Wf + ncol;
    const float* bbp  = Wb + ncol;
    const float* arow = &xs[l15 * LDSP];

    v8f accF = {}, accB = {};
#pragma unroll 4
    for (int k0 = 0; k0 < DIN; k0 += 4) {
        int ka = k0 + khalf;
        v2f a, bf, bb;
        a[0]  = arow[ka];
        a[1]  = arow[ka + 1];
        bf[0] = bfp[(long)ka * HC];
        bf[1] = bfp[(long)(ka + 1) * HC];
        bb[0] = bbp[(long)ka * HC];
        bb[1] = bbp[(long)(ka + 1) * HC];
        accF = __builtin_amdgcn_wmma_f32_16x16x4_f32(false, a, false, bf,
                                                     (short)0, accF, false, false);
        accB = __builtin_amdgcn_wmma_f32_16x16x4_f32(false, a, false, bb,
                                                     (short)0, accB, false, false);
    }
    int mbase = mtile * 16 + ((lane >> 4) << 3);
#pragma unroll
    for (int r = 0; r < 8; ++r) {
        xlF[(long)(mbase + r) * HC + ncol] = accF[r];
        xlB[(long)(mbase + r) * HC + ncol] = accB[r];
    }
}

// -------- per-node attention scores: al = <xl_h, a_src_h>, ar = <xl_h, a_dst_h> ----
__global__ void k_attn(const float* __restrict__ xlF, const float* __restrict__ xlB,
                       const float* __restrict__ asF, const float* __restrict__ adF,
                       const float* __restrict__ asB, const float* __restrict__ adB,
                       float* __restrict__ alF, float* __restrict__ arF,
                       float* __restrict__ alB, float* __restrict__ arB, long N) {
    long id = (long)blockIdx.x * blockDim.x + threadIdx.x;
    if (id >= N * HEADS) return;
    long n = id >> 2;
    int  h = (int)(id & 3);
    const float* rf = xlF + n * HC + h * CH;
    const float* rb = xlB + n * HC + h * CH;
    const float* sF = asF + h * CH;  const float* dF = adF + h * CH;
    const float* sB = asB + h * CH;  const float* dB = adB + h * CH;
    float a0 = 0.f, a1 = 0.f, a2 = 0.f, a3 = 0.f;
#pragma unroll 8
    for (int c = 0; c < CH; ++c) {
        float vf = rf[c], vb = rb[c];
        a0 += vf * sF[c];  a1 += vf * dF[c];
        a2 += vb * sB[c];  a3 += vb * dB[c];
    }
    alF[id] = a0;  arF[id] = a1;  alB[id] = a2;  arB[id] = a3;
}

__device__ __forceinline__ void edge_nodes(const int* __restrict__ src,
                                           const int* __restrict__ dst,
                                           long e, long E, int& s, int& d) {
    if (e < E) { s = src[e]; d = dst[e]; }
    else       { s = d = (int)(e - E); }   // self loops appended
}

__device__ __forceinline__ float edge_score(const float* __restrict__ al,
                                            const float* __restrict__ ar,
                                            int s, int d, int h) {
    float t = al[(long)s * HEADS + h] + ar[(long)d * HEADS + h];
    return t > 0.f ? t : NEG_SLOPE * t;    // leaky relu
}

// -------- edge pass 1: segment max into m[dst][h] (ordered-int atomics) --------
__global__ void k_edge_max(const int* __restrict__ src, const int* __restrict__ dst,
                           long E, long Etot,
                           const float* __restrict__ al, const float* __restrict__ ar,
                           int* __restrict__ m) {
    long id = (long)blockIdx.x * blockDim.x + threadIdx.x;
    if (id >= Etot * HEADS) return;
    long e = id >> 2;  int h = (int)(id & 3);
    int s, d;  edge_nodes(src, dst, e, E, s, d);
    float t = edge_score(al, ar, s, d, h);
    atomicMax(m + (long)d * HEADS + h, enc_f(t));
}

// -------- edge pass 2: den[dst][h] += exp(e - m[dst][h]) --------
__global__ void k_edge_den(const int* __restrict__ src, const int* __restrict__ dst,
                           long E, long Etot,
                           const float* __restrict__ al, const float* __restrict__ ar,
                           const int* __restrict__ m, float* __restrict__ den) {
    long id = (long)blockIdx.x * blockDim.x + threadIdx.x;
    if (id >= Etot * HEADS) return;
    long e = id >> 2;  int h = (int)(id & 3);
    int s, d;  edge_nodes(src, dst, e, E, s, d);
    float t = edge_score(al, ar, s, d, h);
    long di = (long)d * HEADS + h;
    float w = __expf(t - dec_f(m[di]));
    atomicAdd(den + di, w);
}

// -------- edge pass 3: agg[dst][h][:] += alpha * xl[src][h][:] --------
__global__ void k_edge_msg(const int* __restrict__ src, const int* __restrict__ dst,
                           long E, long Etot,
                           const float* __restrict__ al, const float* __restrict__ ar,
                           const int* __restrict__ m, const float* __restrict__ den,
                           const float* __restrict__ xl, float* __restrict__ agg) {
    long id = (long)blockIdx.x * blockDim.x + threadIdx.x;
    if (id >= Etot * HEADS) return;
    long e = id >> 2;  int h = (int)(id & 3);
    int s, d;  edge_nodes(src, dst, e, E, s, d);
    float t = edge_score(al, ar, s, d, h);
    long di = (long)d * HEADS + h;
    float w = __expf(t - dec_f(m[di]));
    float alpha = w / (den[di] + 1e-16f);
    const float* xr = xl  + (long)s * HC + h * CH;
    float*       ao = agg + (long)d * HC + h * CH;
#pragma unroll 8
    for (int c = 0; c < CH; ++c)
        atomicAdd(ao + c, xr[c] * alpha);
}

// -------- bias fold: bvec[n] = sum_k bias[k] * W_fuse[k][n]  (bias = [b_f|b_b]) ----
__global__ void k_bvec(const float* __restrict__ bF, const float* __restrict__ bB,
                       const float* __restrict__ Wfu, float* __restrict__ bvec) {
    int col = blockIdx.x * blockDim.x + threadIdx.x;
    if (col >= HID) return;
    float s = 0.f;
    for (int k = 0; k < HC; ++k) s += bF[k] * Wfu[(long)k * HID + col];
    for (int k = 0; k < HC; ++k) s += bB[k] * Wfu[(long)(HC + k) * HID + col];
    bvec[col] = s;
}

// -------- fusion GEMM: [aggF | aggB] @ W_fuse (+bvec +b_fuse), BN(eval), ReLU ----
// One block = one 16-row M tile; A async-staged to LDS; wave w computes col
// tiles w and w+8 (16 col-tiles total for HID=256).
__global__ void __launch_bounds__(256)
k_fuse_wmma(const float* __restrict__ aggF, const float* __restrict__ aggB,
            const float* __restrict__ Wfu,  const float* __restrict__ bvec,
            const float* __restrict__ bfu,  const float* __restrict__ gamma,
            const float* __restrict__ beta, const float* __restrict__ mean,
            const float* __restrict__ var,  float* __restrict__ out) {
    __shared__ float as_[16 * LDSP];
    const int tid  = threadIdx.x;
    const int lane = tid & 31;
    const int wave = tid >> 5;
    const int mtile = blockIdx.x;

    // stage A = concat(aggF, aggB) rows : 16 x 256 floats
#pragma unroll
    for (int i = 0; i < 4; ++i) {
        int c    = tid + 256 * i;
        int row  = c >> 6;
        int kpos = (c & 63) << 2;
        const float* g = (kpos < HC)
            ? aggF + (long)(mtile * 16 + row) * HC + kpos
            : aggB + (long)(mtile * 16 + row) * HC + (kpos - HC);
        async_g2l_b128(g, &as_[row * LDSP + kpos]);
    }
    async_wait0();
    __syncthreads();

    const int l15   = lane & 15;
    const int khalf = (lane >> 4) << 1;
    const int col0  = wave * 16 + l15;       // cols [0,128)
    const int col1  = col0 + HC;             // cols [128,256)
    const float* b0 = Wfu + col0;
    const float* b1 = Wfu + col1;
    const float* arow = &as_[l15 * LDSP];

    v8f acc0 = {}, acc1 = {};
#pragma unroll 4
    for (int k0 = 0; k0 < HID; k0 += 4) {
        int ka = k0 + khalf;
        v2f a, w0, w1;
        a[0]  = arow[ka];
        a[1]  = arow[ka + 1];
        w0[0] = b0[(long)ka * HID];
        w0[1] = b0[(long)(ka + 1) * HID];
        w1[0] = b1[(long)ka * HID];
        w1[1] = b1[(long)(ka + 1) * HID];
        acc0 = __builtin_amdgcn_wmma_f32_16x16x4_f32(false, a, false, w0,
                                                     (short)0, acc0, false, false);
        acc1 = __builtin_amdgcn_wmma_f32_16x16x4_f32(false, a, false, w1,
                                                     (short)0, acc1, false, false);
    }

    // BN (eval) + ReLU:  relu((acc + bvec + bfu - mean)*gamma*rsqrt(var+eps) + beta)
    const int mbase = mtile * 16 + ((lane >> 4) << 3);
    {
        float sc = gamma[col0] * rsqrtf(var[col0] + 1e-5f);
        float sh = beta[col0] + (bvec[col0] + bfu[col0] - mean[col0]) * sc;
#pragma unroll
        for (int r = 0; r < 8; ++r)
            out[(long)(mbase + r) * HID + col0] = fmaxf(acc0[r] * sc + sh, 0.0f);
    }
    {
        float sc = gamma[col1] * rsqrtf(var[col1] + 1e-5f);
        float sh = beta[col1] + (bvec[col1] + bfu[col1] - mean[col1]) * sc;
#pragma unroll
        for (int r = 0; r < 8; ++r)
            out[(long)(mbase + r) * HID + col1] = fmaxf(acc1[r] * sc + sh, 0.0f);
    }
}

extern "C" void kernel_launch(void* const* d_in, const int* in_sizes, int n_in,
                              void* d_out, int out_size, void* d_ws, size_t ws_size,
                              hipStream_t stream) {
    const float* x       = (const float*)d_in[0];
    const int*   ei      = (const int*)  d_in[1];
    const float* W_f     = (const float*)d_in[2];
    const float* a_src_f = (const float*)d_in[3];
    const float* a_dst_f = (const float*)d_in[4];
    const float* b_f     = (const float*)d_in[5];
    const float* W_b     = (const float*)d_in[6];
    const float* a_src_b = (const float*)d_in[7];
    const float* a_dst_b = (const float*)d_in[8];
    const float* b_b     = (const float*)d_in[9];
    const float* W_fuse  = (const float*)d_in[10];
    const float* b_fuse  = (const float*)d_in[11];
    const float* gamma   = (const float*)d_in[12];
    const float* beta    = (const float*)d_in[13];
    const float* rmean   = (const float*)d_in[14];
    const float* rvar    = (const float*)d_in[15];

    const long N    = in_sizes[0] / DIN;
    const long E    = in_sizes[1] / 2;
    const long Etot = E + N;

    // ---- workspace carve-up (float-sized slots) ----
    float* w    = (float*)d_ws;
    float* xlF  = w;  w += N * HC;
    float* xlB  = w;  w += N * HC;
    float* alF  = w;  w += N * HEADS;
    float* arF  = w;  w += N * HEADS;
    float* alB  = w;  w += N * HEADS;
    float* arB  = w;  w += N * HEADS;
    float* aggF = w;  w += N * HC;      // agg region contiguous (zeroed together)
    float* aggB = w;  w += N * HC;
    float* denF = w;  w += N * HEADS;   // den region contiguous
    float* denB = w;  w += N * HEADS;
    int*   mF   = (int*)w;  w += 2 * N * HEADS;  // m region contiguous (F then B)
    int*   mB   = mF + N * HEADS;
    float* bvec = w;  w += HID;

    // 1) init agg / den / m
    k_init<<<2048, 256, 0, stream>>>(aggF, denF, mF, 2 * N * HC, 2 * N * HEADS);

    // 2) dual projection GEMM (fp32 WMMA, async-LDS staged A)
    const int nMT = (int)((N + 15) / 16);
    k_proj_wmma<<<nMT, 256, 0, stream>>>(x, W_f, W_b, xlF, xlB);

    // 3) attention scores + bias fold vector
    {
        long t = N * HEADS;
        k_attn<<<(int)((t + 255) / 256), 256, 0, stream>>>(
            xlF, xlB, a_src_f, a_dst_f, a_src_b, a_dst_b, alF, arF, alB, arB, N);
        k_bvec<<<1, 256, 0, stream>>>(b_f, b_b, W_fuse, bvec);
    }

    // 4) edge phase, forward (src=ei[0], dst=ei[1]) and backward (reversed edges)
    const int* srcF = ei;      const int* dstF = ei + E;
    const int* srcB = ei + E;  const int* dstB = ei;
    const long tE = Etot * HEADS;
    const int  gE = (int)((tE + 255) / 256);

    k_edge_max<<<gE, 256, 0, stream>>>(srcF, dstF, E, Etot, alF, arF, mF);
    k_edge_den<<<gE, 256, 0, stream>>>(srcF, dstF, E, Etot, alF, arF, mF, denF);
    k_edge_msg<<<gE, 256, 0, stream>>>(srcF, dstF, E, Etot, alF, arF, mF, denF, xlF, aggF);

    k_edge_max<<<gE, 256, 0, stream>>>(srcB, dstB, E, Etot, alB, arB, mB);
    k_edge_den<<<gE, 256, 0, stream>>>(srcB, dstB, E, Etot, alB, arB, mB, denB);
    k_edge_msg<<<gE, 256, 0, stream>>>(srcB, dstB, E, Etot, alB, arB, mB, denB, xlB, aggB);

    // 5) fusion GEMM + BN + ReLU -> d_out
    k_fuse_wmma<<<nMT, 256, 0, stream>>>(aggF, aggB, W_fuse, bvec, b_fuse,
                                         gamma, beta, rmean, rvar, (float*)d_out);
}